// Eq4Net_4423816315258
// MI455X (gfx1250) — compile-verified
//
#include <hip/hip_runtime.h>

typedef __attribute__((ext_vector_type(2))) float v2f;
typedef __attribute__((ext_vector_type(8))) float v8f;

#define NEMBED 1000
#define EDIM   16
#define BB     16
#define NN     24
#define DD     17   // EMBED_DIM + 1 (xfeat channel)

// Single wave (32 lanes). Lane layout: m = lane&15 -> batch, h = lane>>4 -> n-parity.
// Each lane accumulates s[d] = sum over its 12 n-values of relu(embed[xcat[m,n]])[d]
// (and xfeat for d==16), then SWAPX16 ds_swizzle merges the two halves so every
// lane holds the full per-batch sums. s^4 then feeds a chain of 5 fp32 WMMAs
// (16x16x4) computing out[b] = sum_d s4[b,d]*W[d] + bias, K padded 17->20.
__global__ __launch_bounds__(32) void eq4net_kernel(
    const int*   __restrict__ xcat,
    const float* __restrict__ xfeat,
    const float* __restrict__ embed,
    const float* __restrict__ W,
    const float* __restrict__ bias,
    float*       __restrict__ out)
{
    const int lane = threadIdx.x;       // 0..31
    const int m    = lane & 15;         // batch index
    const int h    = lane >> 4;         // 0 or 1 (n parity / K half)

    float s[DD];
#pragma unroll
    for (int d = 0; d < DD; ++d) s[d] = 0.0f;

    // Gather + ReLU + partial sum over n = h, h+2, ..., h+22 (12 rows per lane).
#pragma unroll
    for (int t = 0; t < NN / 2; ++t) {
        const int n   = h + 2 * t;
        const int idx = xcat[m * NN + n];
        const float4* row = (const float4*)(embed + (long)idx * EDIM);
        const float4 r0 = row[0];
        const float4 r1 = row[1];
        const float4 r2 = row[2];
        const float4 r3 = row[3];
        s[0]  += fmaxf(r0.x, 0.0f);  s[1]  += fmaxf(r0.y, 0.0f);
        s[2]  += fmaxf(r0.z, 0.0f);  s[3]  += fmaxf(r0.w, 0.0f);
        s[4]  += fmaxf(r1.x, 0.0f);  s[5]  += fmaxf(r1.y, 0.0f);
        s[6]  += fmaxf(r1.z, 0.0f);  s[7]  += fmaxf(r1.w, 0.0f);
        s[8]  += fmaxf(r2.x, 0.0f);  s[9]  += fmaxf(r2.y, 0.0f);
        s[10] += fmaxf(r2.z, 0.0f);  s[11] += fmaxf(r2.w, 0.0f);
        s[12] += fmaxf(r3.x, 0.0f);  s[13] += fmaxf(r3.y, 0.0f);
        s[14] += fmaxf(r3.z, 0.0f);  s[15] += fmaxf(r3.w, 0.0f);
        s[16] += xfeat[m * NN + n];
    }

    // Cross-half reduction: lane L and L+16 own the same batch, different n's.
    // SWAPX16 swizzle (xor_mask=0x10, and_mask=0x1f) -> both halves now hold
    // the full per-batch sums (duplication is exactly what WMMA A-layout needs).
#pragma unroll
    for (int d = 0; d < DD; ++d) {
        const int other = __builtin_amdgcn_ds_swizzle(__float_as_int(s[d]), 0x401f);
        s[d] += __int_as_float(other);
    }

    // s^4 and W, padded K=17 -> 20 with zeros.
    float s4[20];
    float Wp[20];
#pragma unroll
    for (int d = 0; d < 20; ++d) {
        if (d < DD) {
            const float p = s[d] * s[d];
            s4[d] = p * p;
            Wp[d] = W[d];
        } else {
            s4[d] = 0.0f;
            Wp[d] = 0.0f;
        }
    }

    // C initialized with bias (added exactly once over the chain).
    const float b0 = bias[0];
    v8f acc;
#pragma unroll
    for (int r = 0; r < 8; ++r) acc[r] = b0;

    // A-matrix (16x4 f32) layout: vgpr0 = K=(h?2:0), vgpr1 = K=(h?3:1), M = lane&15.
    // B-matrix (4x16 f32): same K striping; value replicated across the 16 columns.
#pragma unroll
    for (int c = 0; c < 5; ++c) {
        const float a0 = h ? s4[4 * c + 2] : s4[4 * c + 0];
        const float a1 = h ? s4[4 * c + 3] : s4[4 * c + 1];
        const float w0 = h ? Wp[4 * c + 2] : Wp[4 * c + 0];
        const float w1 = h ? Wp[4 * c + 3] : Wp[4 * c + 1];
        v2f A;  A[0] = a0;  A[1] = a1;
        v2f Bv; Bv[0] = w0; Bv[1] = w1;
        acc = __builtin_amdgcn_wmma_f32_16x16x4_f32(
            /*neg_a=*/false, A, /*neg_b=*/false, Bv,
            /*c_mod=*/(short)0, acc, /*reuse_a=*/false, /*reuse_b=*/false);
    }

    // D layout: vgpr r, lanes 0-15 -> M=r, lanes 16-31 -> M=r+8 (all N columns equal).
    // Lane 0 stores out[0..7], lane 16 stores out[8..15].
    if (m == 0) {
#pragma unroll
        for (int r = 0; r < 8; ++r) out[h * 8 + r] = acc[r];
    }
}

extern "C" void kernel_launch(void* const* d_in, const int* in_sizes, int n_in,
                              void* d_out, int out_size, void* d_ws, size_t ws_size,
                              hipStream_t stream) {
    (void)in_sizes; (void)n_in; (void)out_size; (void)d_ws; (void)ws_size;
    const int*   xcat  = (const int*)  d_in[0];
    const float* xfeat = (const float*)d_in[1];
    const float* embed = (const float*)d_in[2];
    const float* W     = (const float*)d_in[3];
    const float* bias  = (const float*)d_in[4];
    float* out = (float*)d_out;
    eq4net_kernel<<<1, 32, 0, stream>>>(xcat, xfeat, embed, W, bias, out);
}